// NODE_74440373174513
// MI455X (gfx1250) — compile-verified
//
#include <hip/hip_runtime.h>

// RK4 step of a 3->8(relu)->1 MLP ODE over 8.4M points.
// Layer-1 GEMM batched onto V_WMMA_F32_16X16X4_F32 (16 points/tile, K=4=[y0,y1,y2,1]).
// Each wave processes 8 tiles; tiles are evaluated in pairs (two independent WMMA
// chains interleave) and stored with one full-wave coalesced store per pair.

typedef float v2f __attribute__((ext_vector_type(2)));
typedef float v8f __attribute__((ext_vector_type(8)));

#define TILES_PER_WAVE 8

static __device__ __forceinline__ float layer2(v8f d, const float w2[8], float bias) {
  // out = b2 + sum_r W2[r] * relu(h[r]); h[r] sits in D-matrix VGPR r of this lane.
  float k = bias;
#pragma unroll
  for (int r = 0; r < 8; ++r) k = fmaf(w2[r], fmaxf(d[r], 0.0f), k);
  return k;
}

// One 16-point tile: y0r = {low: y0, high: y2}, B1v = {low: y1, high: 1}.
// Returns y0 + dy (valid in lanes 0..15).
static __device__ __forceinline__ float rk4_tile(v2f A, float y0r, float B1v,
                                                 bool hi, const float w2[8], float b2s) {
  const v8f zero = {};
  v2f B; B.x = y0r; B.y = B1v;
  v8f d = __builtin_amdgcn_wmma_f32_16x16x4_f32(false, A, false, B, (short)0, zero, false, false);
  const float k1 = layer2(d, w2, b2s);
  B.x = hi ? y0r : fmaf(0.125f, k1, y0r);        // y0 + 0.5*dt*k1, keep y2 in high lanes
  d = __builtin_amdgcn_wmma_f32_16x16x4_f32(false, A, false, B, (short)0, zero, false, false);
  const float k2 = layer2(d, w2, b2s);
  B.x = hi ? y0r : fmaf(0.125f, k2, y0r);        // y0 + 0.5*dt*k2
  d = __builtin_amdgcn_wmma_f32_16x16x4_f32(false, A, false, B, (short)0, zero, false, false);
  const float k3 = layer2(d, w2, b2s);
  B.x = hi ? y0r : fmaf(0.25f, k3, y0r);         // y0 + dt*k3
  d = __builtin_amdgcn_wmma_f32_16x16x4_f32(false, A, false, B, (short)0, zero, false, false);
  const float k4 = layer2(d, w2, b2s);
  return y0r + (k1 + 2.0f * (k2 + k3) + k4) * (0.25f / 6.0f);
}

__global__ __launch_bounds__(256) void node_rk4_wmma(
    const float* __restrict__ x,
    const float* __restrict__ W1p,
    const float* __restrict__ b1p,
    const float* __restrict__ W2p,
    const float* __restrict__ b2p,
    float* __restrict__ out,
    int n)
{
  const int lane = threadIdx.x & 31;
  const int wid  = (blockIdx.x * blockDim.x + threadIdx.x) >> 5;
  const int m    = lane & 15;         // A-row index / B,D-column (element) index
  const bool hi  = (lane >> 4) != 0;  // lane half selects K pair

  const int tile0 = wid * (16 * TILES_PER_WAVE);
  if (tile0 >= n) return;             // wave-uniform

  // ---- A matrix 16x4 = [W1 | b1], rows 8..15 zero.
  // ISA layout: lane = M (mod 16); VGPR0 holds K=2*half, VGPR1 holds K=2*half+1.
  float a0 = 0.0f, a1 = 0.0f;
  if (m < 8) {
    a0 = hi ? W1p[m * 3 + 2] : W1p[m * 3 + 0];   // K=2 : K=0
    a1 = hi ? b1p[m]         : W1p[m * 3 + 1];   // K=3 : K=1
  }
  v2f A; A.x = a0; A.y = a1;

  // ---- layer-2 weights: uniform addresses -> scalar loads (SGPR operands)
  float w2[8];
#pragma unroll
  for (int r = 0; r < 8; ++r) w2[r] = W2p[r];
  const float b2s = b2p[0];

  if (tile0 + 16 * TILES_PER_WAVE <= n) {
    // ------- fast path: branch-free, pair-wise tiles, full-wave stores -------
    // Both per-tile loads are plain immediate-offset loads; the K=1/K=3 row is
    // built with an arithmetic select (B1 = s1*r1 + c1) so the load is consumed
    // unconditionally -> no exec-predicated loads.
    const float* xg0 = x + 3ull * (size_t)tile0 + 3 * m + (hi ? 2 : 0);
    const float* xg1 = xg0 + (hi ? 0 : 1);       // low half: fetches y1; high: y2 again
    const float s1 = hi ? 0.0f : 1.0f;
    const float c1 = hi ? 1.0f : 0.0f;
    float* og = out + (size_t)tile0 + lane;

#pragma unroll
    for (int p = 0; p < TILES_PER_WAVE / 2; ++p) {
      const int tA = 2 * p, tB = 2 * p + 1;
      const float y0a = xg0[48 * tA];
      const float r1a = xg1[48 * tA];
      const float y0b = xg0[48 * tB];
      const float r1b = xg1[48 * tB];
      const float B1a = fmaf(s1, r1a, c1);       // low: y1, high: 1.0 (branchless)
      const float B1b = fmaf(s1, r1b, c1);

      const float ra = rk4_tile(A, y0a, B1a, hi, w2, b2s);   // lanes 0..15 valid
      const float rb = rk4_tile(A, y0b, B1b, hi, w2, b2s);   // lanes 0..15 valid

      // move tile-B results into lanes 16..31, store 32 contiguous floats
      const float rbs = __shfl(rb, m, 32);
      og[32 * p] = hi ? rbs : ra;
    }
  } else {
    // ------- ragged tail: wave-uniform tile guard, clamped loads -------
#pragma unroll 1
    for (int t = 0; t < TILES_PER_WAVE; ++t) {
      const int tile = tile0 + 16 * t;
      if (tile >= n) break;                      // uniform: EXEC all-1s at WMMA
      const int elem = tile + m;
      const int e    = elem < n ? elem : n - 1;  // clamp partial-tile loads
      const size_t base = 3ull * (size_t)e;
      const float y0r = x[base + (hi ? 2 : 0)];
      const float r1  = x[base + 1];
      const float B1v = hi ? 1.0f : r1;
      const float res = rk4_tile(A, y0r, B1v, hi, w2, b2s);
      if (!hi && elem < n) out[elem] = res;
    }
  }
}

extern "C" void kernel_launch(void* const* d_in, const int* in_sizes, int n_in,
                              void* d_out, int out_size, void* d_ws, size_t ws_size,
                              hipStream_t stream)
{
  const float* x  = (const float*)d_in[0];
  const float* W1 = (const float*)d_in[1];
  const float* b1 = (const float*)d_in[2];
  const float* W2 = (const float*)d_in[3];
  const float* b2 = (const float*)d_in[4];
  float* out = (float*)d_out;

  const int n = out_size;                               // B points
  const long long tiles   = ((long long)n + 15) / 16;
  const long long waves   = (tiles + TILES_PER_WAVE - 1) / TILES_PER_WAVE;
  const long long threads = waves * 32;
  const int block = 256;                                // 8 waves / block (wave32)
  const unsigned grid = (unsigned)((threads + block - 1) / block);

  node_rk4_wmma<<<dim3(grid), dim3(block), 0, stream>>>(x, W1, b1, W2, b2, out, n);
}